// GCN2_52819507806386
// MI455X (gfx1250) — compile-verified
//
#include <hip/hip_runtime.h>
#include <hip/hip_bf16.h>
#include <math.h>

typedef __attribute__((ext_vector_type(2))) float v2f;
typedef __attribute__((ext_vector_type(8))) float v8f;

#define HID 128
#define EPSBN 1e-5f

__device__ __forceinline__ void atomic_add_f32(float* p, float v) {
    __hip_atomic_fetch_add(p, v, __ATOMIC_RELAXED, __HIP_MEMORY_SCOPE_AGENT);
}

// ---------------------------------------------------------------------------
// Zero a float region (stats / pooled accumulators)
// ---------------------------------------------------------------------------
__global__ void gcn_zero_f32(float* __restrict__ p, int n) {
    int i = blockIdx.x * blockDim.x + threadIdx.x;
    if (i < n) p[i] = 0.0f;
}

// ---------------------------------------------------------------------------
// Degree: init to 1 (self loop), count edges into col, then rsqrt in place.
// ---------------------------------------------------------------------------
__global__ void gcn_deg_init(float* __restrict__ deg, int n) {
    int i = blockIdx.x * blockDim.x + threadIdx.x;
    if (i < n) deg[i] = 1.0f;
}

__global__ void gcn_deg_count(const long long* __restrict__ col, float* __restrict__ deg, int e) {
    int i = blockIdx.x * blockDim.x + threadIdx.x;
    if (i < e) atomic_add_f32(&deg[(size_t)col[i]], 1.0f);
}

__global__ void gcn_deg_rsqrt(float* __restrict__ deg, int n) {
    int i = blockIdx.x * blockDim.x + threadIdx.x;
    if (i < n) deg[i] = rsqrtf(fmaxf(deg[i], 1.0f));
}

// ---------------------------------------------------------------------------
// WMMA fp32 GEMM:  out[N,128] = act(A[N,K]) @ W[K,128]
// act = identity, or fused BN+ReLU (templated, straight-line inner loop).
// One wave computes a 16x128 stripe via V_WMMA_F32_16X16X4_F32.
// W is staged in LDS K-pair-interleaved: ldsW[(k/2)*PSTRIDE + n*2 + (k&1)]
// so each B operand is one contiguous ds_load_b64 into an even VGPR pair.
// PSTRIDE = 128*2 + 2 pads adjacent pair-rows onto different banks.
// All 8 B pairs of a K-step are loaded into registers before the 8 WMMAs so
// the DS latency is covered by in-flight loads instead of dscnt==0 waits.
// n_rows must be a multiple of 16 (50000 = 3125*16).
// ---------------------------------------------------------------------------
#define PSTRIDE 258

template <bool BN>
__global__ void gcn_gemm_wmma_f32(const float* __restrict__ A, const float* __restrict__ W,
                                  float* __restrict__ out, int n_rows, int K,
                                  const float* __restrict__ scale,
                                  const float* __restrict__ shift) {
    extern __shared__ float ldsW[];  // [(K/2)][PSTRIDE]

    // Stage W (K x 128) into LDS, interleaving consecutive-K pairs per column.
    for (int i = threadIdx.x; i < K * HID; i += blockDim.x) {
        int k = i >> 7, n = i & 127;
        ldsW[(k >> 1) * PSTRIDE + n * 2 + (k & 1)] = W[i];
    }
    __syncthreads();

    const int wave = threadIdx.x >> 5;
    const int lane = threadIdx.x & 31;
    const int lh   = lane & 15;   // A: row M = mtile+lh ; B/D: column N index
    const int lg   = lane >> 4;   // K half-select / D row-half select

    const int mtile = (blockIdx.x * (blockDim.x >> 5) + wave) * 16;
    if (mtile >= n_rows) return;  // wave-uniform branch; EXEC stays all-ones

    v8f acc[8];
#pragma unroll
    for (int n = 0; n < 8; ++n) acc[n] = (v8f){0.f, 0.f, 0.f, 0.f, 0.f, 0.f, 0.f, 0.f};

    const float* arow = A + (size_t)(mtile + lh) * K;

    for (int k0 = 0; k0 < K; k0 += 4) {
        const int ka = k0 + 2 * lg;           // even -> 8B aligned loads
        v2f a = *(const v2f*)(arow + ka);
        if (BN) {  // fused BatchNorm + ReLU on the activation operand
            v2f sc = *(const v2f*)(scale + ka);
            v2f sh = *(const v2f*)(shift + ka);
            a.x = fmaxf(fmaf(a.x, sc.x, sh.x), 0.0f);
            a.y = fmaxf(fmaf(a.y, sc.y, sh.y), 0.0f);
        }
        const float* bbase = ldsW + (size_t)(ka >> 1) * PSTRIDE + lh * 2;
        v2f b[8];
#pragma unroll
        for (int n = 0; n < 8; ++n) {
            b[n] = *(const v2f*)(bbase + n * 32);  // ds_load_b64, pair-contiguous
        }
#pragma unroll
        for (int n = 0; n < 8; ++n) {
            acc[n] = __builtin_amdgcn_wmma_f32_16x16x4_f32(
                false, a, false, b[n], (short)0, acc[n], false, false);
        }
    }

#pragma unroll
    for (int n = 0; n < 8; ++n) {
#pragma unroll
        for (int v = 0; v < 8; ++v) {
            out[(size_t)(mtile + v + 8 * lg) * HID + n * 16 + lh] = acc[n][v];
        }
    }
}

// ---------------------------------------------------------------------------
// agg[i,:] = dinv[i]^2 * xw[i,:] + bias  (self-loop term + bias, init pass)
// ---------------------------------------------------------------------------
__global__ void gcn_agg_init(const float* __restrict__ xw, const float* __restrict__ dinv,
                             const float* __restrict__ bias, float* __restrict__ agg,
                             int total) {
    int i = blockIdx.x * blockDim.x + threadIdx.x;
    if (i < total) {
        int row = i >> 7, f = i & 127;
        float di = dinv[row];
        agg[i] = di * di * xw[i] + bias[f];
    }
}

// ---------------------------------------------------------------------------
// Edge scatter: agg[col,:] += dinv[row]*dinv[col] * xw[row,:]
// One wave per edge; each lane owns a float4 of the 128 features.
// ---------------------------------------------------------------------------
__global__ void gcn_edge_agg(const long long* __restrict__ rows, const long long* __restrict__ cols,
                             const float* __restrict__ dinv, const float* __restrict__ xw,
                             float* __restrict__ agg, int e) {
    const int lane   = threadIdx.x & 31;
    const int warp   = (blockIdx.x * blockDim.x + threadIdx.x) >> 5;
    const int nwarps = (gridDim.x * blockDim.x) >> 5;
    for (int ed = warp; ed < e; ed += nwarps) {
        const long long r = rows[ed];
        const long long c = cols[ed];
        const float norm = dinv[r] * dinv[c];
        const float4 v = *(const float4*)(xw + (size_t)r * HID + lane * 4);
        float* dst = agg + (size_t)c * HID + lane * 4;
        atomic_add_f32(dst + 0, norm * v.x);
        atomic_add_f32(dst + 1, norm * v.y);
        atomic_add_f32(dst + 2, norm * v.z);
        atomic_add_f32(dst + 3, norm * v.w);
    }
}

// ---------------------------------------------------------------------------
// BatchNorm statistics: per-column sum & sum-of-squares (128 cols).
// Block = 128 threads (thread == column), block-strided over rows.
// ---------------------------------------------------------------------------
__global__ void gcn_bn_stats(const float* __restrict__ h, float* __restrict__ sums,
                             float* __restrict__ sumsq, int n) {
    const int c = threadIdx.x;
    float s = 0.f, sq = 0.f;
    for (int r = blockIdx.x; r < n; r += gridDim.x) {
        float v = h[(size_t)r * HID + c];
        s += v;
        sq += v * v;
    }
    atomic_add_f32(&sums[c], s);
    atomic_add_f32(&sumsq[c], sq);
}

__global__ void gcn_bn_finalize(const float* __restrict__ sums, const float* __restrict__ sumsq,
                                const float* __restrict__ g, const float* __restrict__ be,
                                float* __restrict__ scale, float* __restrict__ shift, int n) {
    const int c = threadIdx.x;  // 128 threads
    const float inv_n = 1.0f / (float)n;
    const float mean = sums[c] * inv_n;
    const float var  = fmaxf(sumsq[c] * inv_n - mean * mean, 0.0f);
    const float sc   = g[c] * rsqrtf(var + EPSBN);
    scale[c] = sc;
    shift[c] = be[c] - mean * sc;
}

// ---------------------------------------------------------------------------
// Pooling: apply BN2+ReLU on the fly, segment-sum rows into psum[g,128], pcnt[g]
// ---------------------------------------------------------------------------
__global__ void gcn_pool(const float* __restrict__ h, const float* __restrict__ scale,
                         const float* __restrict__ shift, const long long* __restrict__ batch,
                         float* __restrict__ psum, float* __restrict__ pcnt, int n) {
    const int c = threadIdx.x;  // 128 threads
    for (int r = blockIdx.x; r < n; r += gridDim.x) {
        const long long gidx = batch[r];
        const float v = fmaxf(fmaf(h[(size_t)r * HID + c], scale[c], shift[c]), 0.0f);
        atomic_add_f32(&psum[(size_t)gidx * HID + c], v);
        if (c == 0) atomic_add_f32(&pcnt[gidx], 1.0f);
    }
}

// ---------------------------------------------------------------------------
// Classifier: out[g,c] = sigmoid((psum[g,:]/cnt[g]) @ Wc[:,c] + bc[c]) ; G*C = 3200
// ---------------------------------------------------------------------------
__global__ void gcn_classifier(const float* __restrict__ psum, const float* __restrict__ pcnt,
                               const float* __restrict__ Wc, const float* __restrict__ bc,
                               float* __restrict__ out, int G, int C) {
    int idx = blockIdx.x * blockDim.x + threadIdx.x;
    if (idx < G * C) {
        int g = idx / C, c = idx % C;
        const float inv_cnt = 1.0f / fmaxf(pcnt[g], 1.0f);
        float acc = bc[c];
        const float* prow = psum + (size_t)g * HID;
        for (int k = 0; k < HID; ++k) acc = fmaf(prow[k] * inv_cnt, Wc[k * C + c], acc);
        out[idx] = 1.0f / (1.0f + expf(-acc));
    }
}

// ---------------------------------------------------------------------------
// Launch
// ---------------------------------------------------------------------------
extern "C" void kernel_launch(void* const* d_in, const int* in_sizes, int n_in,
                              void* d_out, int out_size, void* d_ws, size_t ws_size,
                              hipStream_t stream) {
    const float* x   = (const float*)d_in[0];
    const float* W1  = (const float*)d_in[1];
    const float* b1  = (const float*)d_in[2];
    const float* g1  = (const float*)d_in[3];
    const float* be1 = (const float*)d_in[4];
    const float* W2  = (const float*)d_in[5];
    const float* b2  = (const float*)d_in[6];
    const float* g2  = (const float*)d_in[7];
    const float* be2 = (const float*)d_in[8];
    const float* Wc  = (const float*)d_in[9];
    const float* bc  = (const float*)d_in[10];
    const long long* edge_index = (const long long*)d_in[11];
    const long long* batch      = (const long long*)d_in[12];

    const int FIN = 96, G = 64, C = 50;
    const int N = in_sizes[0] / FIN;        // 50000
    const int E = in_sizes[11] / 2;         // 800000
    const long long* erow = edge_index;     // edge_index[0]
    const long long* ecol = edge_index + E; // edge_index[1]

    float* ws = (float*)d_ws;
    const size_t feat = (size_t)N * HID;
    float* bufA  = ws;               // xw of current layer
    float* bufB  = bufA + feat;      // aggregated / activation of current layer
    float* dinv  = bufB + feat;      // degree then dinv, N floats
    float* zbase = dinv + N;         // start of zero-initialized accumulator region
    float* sums1  = zbase;           // 128
    float* sumsq1 = zbase + 128;
    float* sums2  = zbase + 256;
    float* sumsq2 = zbase + 384;
    float* psum   = zbase + 512;         // G*HID = 8192
    float* pcnt   = psum + (size_t)G * HID;  // 64
    const int nzero = 512 + G * HID + G;
    float* scale1 = pcnt + G;        // 4 x 128 fused-BN params
    float* shift1 = scale1 + HID;
    float* scale2 = shift1 + HID;
    float* shift2 = scale2 + HID;

    float* out = (float*)d_out;

    // 0) reset accumulators (deterministic across graph replays)
    gcn_zero_f32<<<(nzero + 255) / 256, 256, 0, stream>>>(zbase, nzero);

    // 1) symmetric-norm degrees (with self loops)
    gcn_deg_init<<<(N + 255) / 256, 256, 0, stream>>>(dinv, N);
    gcn_deg_count<<<(E + 255) / 256, 256, 0, stream>>>(ecol, dinv, E);
    gcn_deg_rsqrt<<<(N + 255) / 256, 256, 0, stream>>>(dinv, N);

    const int mtiles = N / 16;
    const int gblocks = (mtiles + 7) / 8;

    // 2) layer 1: xw = x @ W1  (fp32 WMMA), then self-loop+bias init, edge scatter
    gcn_gemm_wmma_f32<false><<<gblocks, 256, (FIN / 2) * PSTRIDE * sizeof(float), stream>>>(
        x, W1, bufA, N, FIN, nullptr, nullptr);
    gcn_agg_init<<<((int)feat + 255) / 256, 256, 0, stream>>>(bufA, dinv, b1, bufB, (int)feat);
    gcn_edge_agg<<<4096, 256, 0, stream>>>(erow, ecol, dinv, bufA, bufB, E);

    // 3) BN1 stats -> fused scale/shift (BN+ReLU applied inside layer-2 GEMM)
    gcn_bn_stats<<<512, HID, 0, stream>>>(bufB, sums1, sumsq1, N);
    gcn_bn_finalize<<<1, HID, 0, stream>>>(sums1, sumsq1, g1, be1, scale1, shift1, N);

    // 4) layer 2: xw2 = relu(bn(h1)) @ W2 (fused), self-loop+bias, edge scatter
    gcn_gemm_wmma_f32<true><<<gblocks, 256, (HID / 2) * PSTRIDE * sizeof(float), stream>>>(
        bufB, W2, bufA, N, HID, scale1, shift1);
    gcn_agg_init<<<((int)feat + 255) / 256, 256, 0, stream>>>(bufA, dinv, b2, bufB, (int)feat);
    gcn_edge_agg<<<4096, 256, 0, stream>>>(erow, ecol, dinv, bufA, bufB, E);

    // 5) BN2 stats -> fused scale/shift (applied inside pooling)
    gcn_bn_stats<<<512, HID, 0, stream>>>(bufB, sums2, sumsq2, N);
    gcn_bn_finalize<<<1, HID, 0, stream>>>(sums2, sumsq2, g2, be2, scale2, shift2, N);

    // 6) global mean pool (BN2+ReLU fused) + classifier + sigmoid
    gcn_pool<<<1024, HID, 0, stream>>>(bufB, scale2, shift2, batch, psum, pcnt, N);
    gcn_classifier<<<(G * C + 255) / 256, 256, 0, stream>>>(psum, pcnt, Wc, bc, out, G, C);
}